// AdditiveAttention_31310311587851
// MI455X (gfx1250) — compile-verified
//
#include <hip/hip_runtime.h>
#include <hip/hip_bf16.h>

typedef __attribute__((ext_vector_type(2))) float v2f;
typedef __attribute__((ext_vector_type(8))) float v8f;

#define WMMA_F32(a, b, c) \
  __builtin_amdgcn_wmma_f32_16x16x4_f32(false, (a), false, (b), (short)0, (c), false, false)

// ---------------------------------------------------------------------------
// Y (rows x N) = X (rows x K) @ W (K x N), all row-major fp32.
// One wave32 per 16x16 output tile; K-loop of V_WMMA_F32_16X16X4_F32.
// A layout (16x4 f32): VGPR v, lane L -> A[m = L%16][k = v + 2*(L>=16)]
// B layout (4x16 f32): VGPR v, lane L -> B[k = v + 2*(L>=16)][n = L%16]
// D layout (16x16 f32): VGPR v, lane L -> D[m = v + 8*(L>=16)][n = L%16]
// ---------------------------------------------------------------------------
__global__ void __launch_bounds__(32)
proj_kernel(const float* __restrict__ X, const float* __restrict__ W,
            float* __restrict__ Y, int K, int N) {
  const int lane = threadIdx.x;
  const int col  = lane & 15;
  const int kh   = lane >> 4;           // 0 or 1
  const int row0 = blockIdx.x * 16;
  const int n0   = blockIdx.y * 16;

  v8f c = {};
  const float* xrow = X + (size_t)(row0 + col) * K;
  for (int k0 = 0; k0 < K; k0 += 4) {
    v2f a, b;
    a.x = xrow[k0 + 2 * kh];
    a.y = xrow[k0 + 1 + 2 * kh];
    b.x = W[(size_t)(k0 + 2 * kh) * N + n0 + col];
    b.y = W[(size_t)(k0 + 1 + 2 * kh) * N + n0 + col];
    c = WMMA_F32(a, b, c);
  }
#pragma unroll
  for (int v = 0; v < 8; ++v)
    Y[(size_t)(row0 + v + 8 * kh) * N + n0 + col] = c[v];
}

// ---------------------------------------------------------------------------
// Fused: scores = wv . tanh(qp[q] + kp[k])  -> masked softmax -> attn @ V
// One 256-thread workgroup per (batch, 16-row q tile).  B=8, Tq=128, Tk=512,
// D=H=256.  LDS: qp tile 16x256 f32 (16KB) + scores 16x512 f32 (32KB) + wv.
// ---------------------------------------------------------------------------
__global__ void __launch_bounds__(256)
attn_kernel(const float* __restrict__ qp, const float* __restrict__ kp,
            const float* __restrict__ V, const int* __restrict__ valid_lens,
            const float* __restrict__ wv, float* __restrict__ out) {
  constexpr int Tq = 128, Tk = 512, D = 256, H = 256;
  const int b  = blockIdx.x >> 3;          // Tq/16 == 8 tiles per batch
  const int qt = blockIdx.x & 7;
  const int q0 = qt * 16;
  const int t  = threadIdx.x;

  __shared__ float s_qp[16 * H];           // 16 KB
  __shared__ float s_sc[16 * Tk];          // 32 KB  (scores -> attn probs)
  __shared__ float s_wv[H];                // 1 KB

  // ---- stage qp tile + wv into LDS ----
  const float* qpb = qp + (size_t)(b * Tq + q0) * H;
  for (int i = t; i < 16 * H; i += 256) s_qp[i] = qpb[i];
  for (int i = t; i < H; i += 256)      s_wv[i] = wv[i];
  __syncthreads();

  // ---- scores: 16*512 entries, each a 256-deep tanh reduction ----
  const float* kpb = kp + (size_t)b * Tk * H;
  for (int e = t; e < 16 * Tk; e += 256) {
    const int q = e >> 9;                  // Tk = 512
    const int k = e & (Tk - 1);
    const float* qrow = s_qp + q * H;
    const float* krow = kpb + (size_t)k * H;
    float s = 0.f;
    for (int h = 0; h < H; ++h)
      s += s_wv[h] * tanhf(qrow[h] + krow[h]);
    s_sc[q * Tk + k] = s;
  }
  __syncthreads();

  // ---- masked softmax, one row per thread (16 rows) ----
  if (t < 16) {
    int vl = valid_lens[b];
    if (vl > Tk) vl = Tk;
    float* row = s_sc + t * Tk;
    if (vl <= 0) {
      for (int k = 0; k < Tk; ++k) row[k] = 0.f;   // fully masked -> zeros
    } else {
      float m = -3.402823466e38f;
      for (int k = 0; k < vl; ++k) m = fmaxf(m, row[k]);
      float sum = 0.f;
      for (int k = 0; k < Tk; ++k) {
        float p = (k < vl) ? __expf(row[k] - m) : 0.f;
        row[k] = p;
        sum += p;
      }
      const float inv = 1.f / sum;
      for (int k = 0; k < Tk; ++k) row[k] *= inv;
    }
  }
  __syncthreads();

  // ---- out tile (16 x 256) = attn (16 x 512) @ V_b (512 x 256) via WMMA ----
  const int wave = t >> 5;                 // 8 waves, 2 N-tiles each
  const int lane = t & 31;
  const int col  = lane & 15;
  const int kh   = lane >> 4;
  const float* Vb = V + (size_t)b * Tk * D;
  for (int nt = wave * 2; nt < wave * 2 + 2; ++nt) {
    const int n0 = nt * 16;
    v8f c = {};
    for (int k0 = 0; k0 < Tk; k0 += 4) {
      v2f a, bb;
      a.x  = s_sc[col * Tk + k0 + 2 * kh];            // A[m=col][k]
      a.y  = s_sc[col * Tk + k0 + 1 + 2 * kh];
      bb.x = Vb[(size_t)(k0 + 2 * kh) * D + n0 + col];     // B[k][n=col]
      bb.y = Vb[(size_t)(k0 + 1 + 2 * kh) * D + n0 + col];
      c = WMMA_F32(a, bb, c);
    }
#pragma unroll
    for (int v = 0; v < 8; ++v)
      out[(size_t)(b * Tq + q0 + v + 8 * kh) * D + n0 + col] = c[v];
  }
}

extern "C" void kernel_launch(void* const* d_in, const int* in_sizes, int n_in,
                              void* d_out, int out_size, void* d_ws, size_t ws_size,
                              hipStream_t stream) {
  constexpr int B = 8, Tq = 128, Tk = 512, D = 256, H = 256;
  const float* queries = (const float*)d_in[0];  // (B,Tq,D)
  const float* keys    = (const float*)d_in[1];  // (B,Tk,D)
  const float* values  = (const float*)d_in[2];  // (B,Tk,D)
  const int*   vlens   = (const int*)d_in[3];    // (B,)
  const float* Wq      = (const float*)d_in[4];  // (D,H)
  const float* Wk      = (const float*)d_in[5];  // (D,H)
  const float* wv      = (const float*)d_in[6];  // (H,)
  float* out = (float*)d_out;                    // (B,Tq,D)

  float* qp = (float*)d_ws;                      // (B*Tq, H) = 1 MB
  float* kp = qp + (size_t)B * Tq * H;           // (B*Tk, H) = 4 MB

  // q_proj = Q @ Wq   (1024 x 256) @ (256 x 256)
  proj_kernel<<<dim3((B * Tq) / 16, H / 16), 32, 0, stream>>>(queries, Wq, qp, D, H);
  // k_proj = K @ Wk   (4096 x 256) @ (256 x 256)
  proj_kernel<<<dim3((B * Tk) / 16, H / 16), 32, 0, stream>>>(keys, Wk, kp, D, H);
  // fused scores + masked softmax + attn @ V
  attn_kernel<<<B * (Tq / 16), 256, 0, stream>>>(qp, kp, values, vlens, wv, out);
}